// ExpertsModule_60550448939675
// MI455X (gfx1250) — compile-verified
//
#include <hip/hip_runtime.h>
#include <hip/hip_bf16.h>
#include <stdint.h>

// ---------------------------------------------------------------------------
// MoE Phi3-MLP (f32) for gfx1250 using native f32 WMMA (V_WMMA_F32_16X16X4_F32)
//   phase 1: act[e,t,i] = up * silu(gate),  gu = X @ Wgu_e      (K = 1024)
//   phase 2: out[t,h]   = sum_e mask[t,e] * (act_e @ Wd_e)[t,h] (K = 4096)
// A-tiles staged into LDS via CDNA5 async global->LDS DMA (ASYNCcnt path).
// ---------------------------------------------------------------------------

typedef __attribute__((ext_vector_type(2))) float v2f;
typedef __attribute__((ext_vector_type(8))) float v8f;

#define TOKENS 1024
#define HIDDEN 1024
#define INTER  4096
#define NEXP   8
#define KC     64      // K-chunk staged in LDS per iteration

__device__ __forceinline__ v8f wmma_f32(v2f a, v2f b, v8f c) {
  // D = A(16x4,f32) * B(4x16,f32) + C(16x16,f32)
  return __builtin_amdgcn_wmma_f32_16x16x4_f32(
      /*neg_a=*/false, a, /*neg_b=*/false, b,
      /*c_mod=*/(short)0, c, /*reuse_a=*/false, /*reuse_b=*/false);
}

// CDNA5 async global -> LDS copy, 16 bytes per lane, tracked with ASYNCcnt.
// lds_byte_off: LDS byte address (low 32 bits of the generic shared pointer).
__device__ __forceinline__ void async_copy_b128(uint32_t lds_byte_off,
                                                const void* gaddr) {
  asm volatile("global_load_async_to_lds_b128 %0, %1, off"
               :
               : "v"(lds_byte_off), "v"(gaddr)
               : "memory");
}

__device__ __forceinline__ void wait_async0() {
  asm volatile("s_wait_asynccnt 0x0" ::: "memory");
}

// ---------------------------------------------------------------------------
// Phase 1: gate_up GEMM + SiLU-gate epilogue.
// Grid: (INTER/64, TOKENS/16, NEXP), block = 128 (4 waves).
// Each wave owns one 16x16 inter-column tile (gate + matching up column).
// ---------------------------------------------------------------------------
__global__ __launch_bounds__(128) void moe_gateup_silu(
    const float* __restrict__ X,    // [TOKENS, HIDDEN]
    const float* __restrict__ Wgu,  // [NEXP, HIDDEN, 2*INTER]
    float* __restrict__ act)        // [NEXP, TOKENS, INTER]
{
  __shared__ __align__(16) float ldsX[16 * KC];

  const int tid  = threadIdx.x;
  const int lane = tid & 31;
  const int wave = tid >> 5;      // 0..3
  const int g    = lane >> 4;     // half-wave group (0/1)
  const int n    = lane & 15;     // N column / A row within half-wave

  const int e  = blockIdx.z;
  const int t0 = blockIdx.y * 16;
  const int j0 = (blockIdx.x * 4 + wave) * 16;  // inter-column tile base

  const float* wg = Wgu + (size_t)e * HIDDEN * (2 * INTER);
  const uint32_t lds_base = (uint32_t)(uintptr_t)&ldsX[0];

  v8f cg = {};   // gate accumulator (16x16 f32)
  v8f cu = {};   // up   accumulator (16x16 f32)

  for (int k0 = 0; k0 < HIDDEN; k0 += KC) {
    // Async-stage X[t0:t0+16, k0:k0+KC) into LDS: 256 float4s, 2 per thread.
    {
      const int i0 = tid;            // float4 index, 16 float4 per row
      const int r0 = i0 >> 4, c0 = i0 & 15;
      const int i1 = tid + 128;
      const int r1 = i1 >> 4, c1 = i1 & 15;
      async_copy_b128(lds_base + (uint32_t)i0 * 16,
                      X + (size_t)(t0 + r0) * HIDDEN + (k0 + c0 * 4));
      async_copy_b128(lds_base + (uint32_t)i1 * 16,
                      X + (size_t)(t0 + r1) * HIDDEN + (k0 + c1 * 4));
      wait_async0();
    }
    __syncthreads();

#pragma unroll
    for (int kk = 0; kk < KC; kk += 4) {
      const int kr = kk + 2 * g;                      // K for VGPR pair {kr, kr+1}
      const v2f a = *(const v2f*)&ldsX[n * KC + kr];  // A[m=n, kr..kr+1]

      const float* bp = wg + (size_t)(k0 + kr) * (2 * INTER) + (j0 + n);
      v2f bg, bu;
      bg.x = bp[0];                     // gate col, K=kr
      bg.y = bp[2 * INTER];             // gate col, K=kr+1
      bu.x = bp[INTER];                 // up col (offset +INTER in fused weight)
      bu.y = bp[2 * INTER + INTER];
      cg = wmma_f32(a, bg, cg);
      cu = wmma_f32(a, bu, cu);
    }
    __syncthreads();
  }

  // Epilogue: act = up * silu(gate); VGPR v holds row M = v + 8*g, col n.
  float* ap = act + ((size_t)e * TOKENS + t0) * INTER + (j0 + n);
#pragma unroll
  for (int v = 0; v < 8; ++v) {
    const int row = v + 8 * g;
    const float gate = cg[v];
    const float up   = cu[v];
    const float sig  = 1.0f / (1.0f + __expf(-gate));
    ap[(size_t)row * INTER] = up * gate * sig;
  }
}

// ---------------------------------------------------------------------------
// Phase 2: down-proj GEMM per expert, masked accumulate over experts.
// Grid: (HIDDEN/64, TOKENS/16), block = 128 (4 waves).
// Each wave owns one 16x16 hidden tile; experts looped inside.
// ---------------------------------------------------------------------------
__global__ __launch_bounds__(128) void moe_down_mask(
    const float* __restrict__ act,   // [NEXP, TOKENS, INTER]
    const float* __restrict__ Wd,    // [NEXP, INTER, HIDDEN]
    const int*   __restrict__ mask,  // [TOKENS, NEXP] in {0,1}
    float* __restrict__ out)         // [TOKENS, HIDDEN]
{
  __shared__ __align__(16) float ldsA[16 * KC];

  const int tid  = threadIdx.x;
  const int lane = tid & 31;
  const int wave = tid >> 5;
  const int g    = lane >> 4;
  const int n    = lane & 15;

  const int t0 = blockIdx.y * 16;
  const int h0 = (blockIdx.x * 4 + wave) * 16;
  const uint32_t lds_base = (uint32_t)(uintptr_t)&ldsA[0];

  v8f acc = {};

  for (int e = 0; e < NEXP; ++e) {
    v8f c = {};
    const float* ae = act + ((size_t)e * TOKENS + t0) * INTER;
    const float* we = Wd + (size_t)e * INTER * HIDDEN;

    for (int k0 = 0; k0 < INTER; k0 += KC) {
      // Async-stage act[e, t0:t0+16, k0:k0+KC) into LDS.
      {
        const int i0 = tid;
        const int r0 = i0 >> 4, c0 = i0 & 15;
        const int i1 = tid + 128;
        const int r1 = i1 >> 4, c1 = i1 & 15;
        async_copy_b128(lds_base + (uint32_t)i0 * 16,
                        ae + (size_t)r0 * INTER + (k0 + c0 * 4));
        async_copy_b128(lds_base + (uint32_t)i1 * 16,
                        ae + (size_t)r1 * INTER + (k0 + c1 * 4));
        wait_async0();
      }
      __syncthreads();

#pragma unroll
      for (int kk = 0; kk < KC; kk += 4) {
        const int kr = kk + 2 * g;
        const v2f a = *(const v2f*)&ldsA[n * KC + kr];
        const float* bp = we + (size_t)(k0 + kr) * HIDDEN + (h0 + n);
        v2f b;
        b.x = bp[0];
        b.y = bp[HIDDEN];
        c = wmma_f32(a, b, c);
      }
      __syncthreads();
    }

    // Masked accumulate: row M = v + 8*g belongs to token t0 + M.
#pragma unroll
    for (int v = 0; v < 8; ++v) {
      const int row = t0 + v + 8 * g;
      acc[v] += (float)mask[row * NEXP + e] * c[v];
    }
  }

#pragma unroll
  for (int v = 0; v < 8; ++v) {
    out[(size_t)(t0 + v + 8 * g) * HIDDEN + (h0 + n)] = acc[v];
  }
}

// ---------------------------------------------------------------------------
extern "C" void kernel_launch(void* const* d_in, const int* in_sizes, int n_in,
                              void* d_out, int out_size, void* d_ws, size_t ws_size,
                              hipStream_t stream) {
  (void)in_sizes; (void)n_in; (void)out_size; (void)ws_size;

  const float* X   = (const float*)d_in[0];  // hidden_states [1024,1024] f32
  const int*   idx = (const int*)  d_in[1];  // expert_indices [1024,8] i32
  const float* Wgu = (const float*)d_in[2];  // w_gate_up [8,1024,8192] f32
  const float* Wd  = (const float*)d_in[3];  // w_down    [8,4096,1024] f32
  float* out = (float*)d_out;                // [1024,1024] f32
  float* act = (float*)d_ws;                 // needs 8*1024*4096*4 = 128 MiB

  const dim3 blk(128);
  const dim3 g1(INTER / 64, TOKENS / 16, NEXP);
  moe_gateup_silu<<<g1, blk, 0, stream>>>(X, Wgu, act);

  const dim3 g2(HIDDEN / 64, TOKENS / 16, 1);
  moe_down_mask<<<g2, blk, 0, stream>>>(act, Wd, idx, out);
}